// K2Layer_79826262163472
// MI455X (gfx1250) — compile-verified
//
#include <hip/hip_runtime.h>
#include <math.h>

// ---------------- problem constants ----------------
static const int Bb   = 4;
static const int Ww   = 1024;
static const int Dd   = 1024;
static const int Rr   = 32;
static const int BLK  = 128;
static const int NBLK = Ww / BLK;     // 8
static const int ROWS = Bb * Ww;      // 4096

typedef float v2f __attribute__((ext_vector_type(2)));
typedef float v8f __attribute__((ext_vector_type(8)));

#define WMMA4(a, b, c) \
    __builtin_amdgcn_wmma_f32_16x16x4_f32(false, (a), false, (b), (short)0, (c), false, false)

__device__ __forceinline__ float sigmoidf_(float x) { return 1.0f / (1.0f + expf(-x)); }
__device__ __forceinline__ float gammaf_(float dl)  { return 0.9f + 0.1f * sigmoidf_(dl); }
__device__ __forceinline__ float geluf_(float x)    { return 0.5f * x * (1.0f + erff(x * 0.70710678118654752f)); }

// ---------------- 1/8: RMSNorm (one block per row, one float4 per thread) ----------------
__global__ void k2_rmsnorm(const float* __restrict__ x, const float* __restrict__ scale,
                           float* __restrict__ out) {
    const int row = blockIdx.x;
    const int t = threadIdx.x;
    const float4* xr4 = (const float4*)(x + (size_t)row * Dd);
    const float4 vv = xr4[t];
    __shared__ float red[256];
    red[t] = vv.x * vv.x + vv.y * vv.y + vv.z * vv.z + vv.w * vv.w;
    __syncthreads();
    for (int s = 128; s > 0; s >>= 1) {
        if (t < s) red[t] += red[t + s];
        __syncthreads();
    }
    const float inv = rsqrtf(red[0] * (1.0f / (float)Dd) + 1e-8f);
    const float4 sc = ((const float4*)scale)[t];
    float4 o;
    o.x = vv.x * inv * sc.x; o.y = vv.y * inv * sc.y;
    o.z = vv.z * inv * sc.z; o.w = vv.w * inv * sc.w;
    ((float4*)(out + (size_t)row * Dd))[t] = o;
}

// ---------------- 2: q/k low-rank projection + l2norm ----------------
__global__ void k2_qk(const float* __restrict__ hn, const float* __restrict__ u,
                      const float* __restrict__ v, float* __restrict__ q,
                      float* __restrict__ kq) {
    const int row = blockIdx.x;
    const int t = threadIdx.x;           // 128 threads
    const int r = t & 31;
    const int c = t >> 5;                // 4 chunks of 256
    const float* x = hn + (size_t)row * Dd;
    float aq = 0.0f, ak = 0.0f;
    for (int d = c * 256; d < c * 256 + 256; ++d) {
        const float xv = x[d];
        aq += xv * u[(size_t)d * Rr + r];
        ak += xv * v[(size_t)d * Rr + r];
    }
    __shared__ float sq[128], sk[128], nrm[2];
    sq[t] = aq; sk[t] = ak;
    __syncthreads();
    if (t < 32) {
        sq[t] = sq[t] + sq[t + 32] + sq[t + 64] + sq[t + 96];
        sk[t] = sk[t] + sk[t + 32] + sk[t + 64] + sk[t + 96];
    }
    __syncthreads();
    if (t == 0) {
        float nq = 0.0f, nk = 0.0f;
        for (int i = 0; i < 32; ++i) { nq += sq[i] * sq[i]; nk += sk[i] * sk[i]; }
        nrm[0] = fmaxf(sqrtf(nq), 1e-8f);
        nrm[1] = fmaxf(sqrtf(nk), 1e-8f);
    }
    __syncthreads();
    if (t < 32) {
        q [(size_t)row * Rr + t] = sq[t] / nrm[0];
        kq[(size_t)row * Rr + t] = sk[t] / nrm[1];
    }
}

// ---------------- 3: base causal mix  out = g * (kb .* tril) @ h_norm ----------------
// grid (2, 64, B), block 256 (8 waves); wave -> 16(M) x 64(N), K truncated at diagonal.
// Main K-loop is unmasked; only the final 16-wide diagonal band applies a post-load select.
__global__ void k2_base_gemm(const float* __restrict__ kb, const float* __restrict__ hn,
                             const float* __restrict__ gate_logit, float* __restrict__ out) {
    const int b = blockIdx.z;
    const int m0 = blockIdx.y * 16;
    const int wave = threadIdx.x >> 5, lane = threadIdx.x & 31;
    const int half = lane >> 4, lid = lane & 15;
    const int n0 = (blockIdx.x * 8 + wave) * 64;
    const float g = sigmoidf_(gate_logit[0]);
    const float* hb = hn + (size_t)b * Ww * Dd;
    const int row = m0 + lid;
    const float* arow = kb + (size_t)row * Ww;
    v8f acc0 = {0,0,0,0,0,0,0,0}, acc1 = acc0, acc2 = acc0, acc3 = acc0;
    int k = 0;
    for (; k < m0; k += 4) {                       // strictly below diagonal: no mask
        const int ka = k + 2 * half;
        v2f a; a.x = arow[ka]; a.y = arow[ka + 1];
        const float* b0 = hb + (size_t)ka * Dd;
        const float* b1 = hb + (size_t)(ka + 1) * Dd;
        v2f bb;
        bb.x = b0[n0 + lid];      bb.y = b1[n0 + lid];      acc0 = WMMA4(a, bb, acc0);
        bb.x = b0[n0 + 16 + lid]; bb.y = b1[n0 + 16 + lid]; acc1 = WMMA4(a, bb, acc1);
        bb.x = b0[n0 + 32 + lid]; bb.y = b1[n0 + 32 + lid]; acc2 = WMMA4(a, bb, acc2);
        bb.x = b0[n0 + 48 + lid]; bb.y = b1[n0 + 48 + lid]; acc3 = WMMA4(a, bb, acc3);
    }
#pragma unroll
    for (int kk = 0; kk < 16; kk += 4) {           // diagonal band: load then select
        const int ka = m0 + kk + 2 * half;
        const float x0 = arow[ka];
        const float x1 = arow[ka + 1];
        v2f a;
        a.x = (ka     <= row) ? x0 : 0.0f;
        a.y = (ka + 1 <= row) ? x1 : 0.0f;
        const float* b0 = hb + (size_t)ka * Dd;
        const float* b1 = hb + (size_t)(ka + 1) * Dd;
        v2f bb;
        bb.x = b0[n0 + lid];      bb.y = b1[n0 + lid];      acc0 = WMMA4(a, bb, acc0);
        bb.x = b0[n0 + 16 + lid]; bb.y = b1[n0 + 16 + lid]; acc1 = WMMA4(a, bb, acc1);
        bb.x = b0[n0 + 32 + lid]; bb.y = b1[n0 + 32 + lid]; acc2 = WMMA4(a, bb, acc2);
        bb.x = b0[n0 + 48 + lid]; bb.y = b1[n0 + 48 + lid]; acc3 = WMMA4(a, bb, acc3);
    }
    float* ob = out + ((size_t)b * Ww + m0) * Dd;
#pragma unroll
    for (int i = 0; i < 8; ++i) {
        const int m = i + 8 * half;
        ob[(size_t)m * Dd + n0      + lid] = acc0[i] * g;
        ob[(size_t)m * Dd + n0 + 16 + lid] = acc1[i] * g;
        ob[(size_t)m * Dd + n0 + 32 + lid] = acc2[i] * g;
        ob[(size_t)m * Dd + n0 + 48 + lid] = acc3[i] * g;
    }
}

// ---------------- 4: per-block decayed contribution C_j[b,j,r,d] ----------------
// grid (R/2, NBLK, B), block 256; thread owns 2 ranks x one float4 of d
__global__ void k2_cblk(const float* __restrict__ hn, const float* __restrict__ kq,
                        const float* __restrict__ decay_logit, float* __restrict__ cblk) {
    const int b = blockIdx.z, j = blockIdx.y, r0 = blockIdx.x * 2;
    const int t = threadIdx.x;
    const float g0 = gammaf_(decay_logit[r0]);
    const float g1 = gammaf_(decay_logit[r0 + 1]);
    const float* hb = hn + ((size_t)b * Ww + j * BLK) * Dd;
    const float* kb = kq + ((size_t)b * Ww + j * BLK) * Rr;
    float a0x = 0, a0y = 0, a0z = 0, a0w = 0;
    float a1x = 0, a1y = 0, a1z = 0, a1w = 0;
    for (int i = 0; i < BLK; ++i) {
        const float k0 = kb[(size_t)i * Rr + r0];
        const float k1 = kb[(size_t)i * Rr + r0 + 1];
        const float4 hv = ((const float4*)(hb + (size_t)i * Dd))[t];
        a0x = g0 * a0x + k0 * hv.x; a0y = g0 * a0y + k0 * hv.y;
        a0z = g0 * a0z + k0 * hv.z; a0w = g0 * a0w + k0 * hv.w;
        a1x = g1 * a1x + k1 * hv.x; a1y = g1 * a1y + k1 * hv.y;
        a1z = g1 * a1z + k1 * hv.z; a1w = g1 * a1w + k1 * hv.w;
    }
    float* c0 = cblk + (((size_t)b * NBLK + j) * Rr + r0) * Dd;
    float4 s0; s0.x = a0x; s0.y = a0y; s0.z = a0z; s0.w = a0w;
    float4 s1; s1.x = a1x; s1.y = a1y; s1.z = a1z; s1.w = a1w;
    ((float4*)c0)[t]        = s0;
    ((float4*)(c0 + Dd))[t] = s1;
}

// ---------------- 5: sequential 8-block state prefix scan ----------------
__global__ void k2_scan(const float* __restrict__ cblk, const float* __restrict__ decay_logit,
                        float* __restrict__ sstart) {
    const int idx = blockIdx.x * 256 + threadIdx.x;   // < B*R*D
    const int d = idx & (Dd - 1);
    const int r = (idx >> 10) & (Rr - 1);
    const int b = idx >> 15;
    const float g  = gammaf_(decay_logit[r]);
    const float gL = powf(g, (float)BLK);
    float s = 0.0f;
    for (int j = 0; j < NBLK; ++j) {
        const size_t o = (((size_t)b * NBLK + j) * Rr + r) * Dd + d;
        sstart[o] = s;
        s = gL * s + cblk[o];
    }
}

// ---------------- 6a: decayed causal scores + q' -> packed A2 [128 x 160] ----------------
// grid (NBLK, B), block 256 (8 waves)
__global__ void k2_scores(const float* __restrict__ q, const float* __restrict__ kq,
                          const float* __restrict__ decay_logit, float* __restrict__ a2) {
    const int blk = blockIdx.x, b = blockIdx.y;
    __shared__ float pwp[128 * 32];
    __shared__ float pwm[128 * 32];
    const int t = threadIdx.x;
    for (int i = t; i < 128 * 32; i += 256) {
        const int l = i >> 5, r = i & 31;
        const float g = gammaf_(decay_logit[r]);
        const float p = powf(g, (float)l);
        pwp[i] = p;
        pwm[i] = 1.0f / p;
    }
    __syncthreads();
    const int wave = t >> 5, lane = t & 31, half = lane >> 4, lid = lane & 15;
    const float* qb = q  + ((size_t)b * Ww + blk * BLK) * Rr;
    const float* kb = kq + ((size_t)b * Ww + blk * BLK) * Rr;
    float* a2b = a2 + ((size_t)(b * NBLK + blk)) * 128 * 160;
    for (int tid = wave; tid < 64; tid += 8) {
        const int mt = tid >> 3, nt = tid & 7;
        if (nt > mt) {
#pragma unroll
            for (int i = 0; i < 8; ++i) {
                const int m = mt * 16 + i + 8 * half, n = nt * 16 + lid;
                a2b[(size_t)m * 160 + n] = 0.0f;
            }
            continue;
        }
        v8f acc = {0,0,0,0,0,0,0,0};
        const int mrow = mt * 16 + lid;
        const int ncol = nt * 16 + lid;
#pragma unroll
        for (int k = 0; k < 32; k += 4) {
            const int ka = k + 2 * half;
            v2f a, bb;
            a.x  = qb[(size_t)mrow * Rr + ka    ] * pwp[mrow * 32 + ka];
            a.y  = qb[(size_t)mrow * Rr + ka + 1] * pwp[mrow * 32 + ka + 1];
            bb.x = kb[(size_t)ncol * Rr + ka    ] * pwm[ncol * 32 + ka];
            bb.y = kb[(size_t)ncol * Rr + ka + 1] * pwm[ncol * 32 + ka + 1];
            acc = WMMA4(a, bb, acc);
        }
#pragma unroll
        for (int i = 0; i < 8; ++i) {
            const int m = mt * 16 + i + 8 * half, n = nt * 16 + lid;
            a2b[(size_t)m * 160 + n] = (n <= m) ? acc[i] : 0.0f;
        }
    }
    // q' = q * gamma^{i+1} into columns 128..159
    for (int i = t; i < 128 * 32; i += 256) {
        const int l = i >> 5, r = i & 31;
        const float g = gammaf_(decay_logit[r]);
        a2b[(size_t)l * 160 + 128 + r] = qb[(size_t)l * Rr + r] * pwp[l * 32 + r] * g;
    }
}

// ---------------- 6b: y = A2 @ [h_blk ; S_start], mix += alpha*y ----------------
// grid (D/128, NBLK, B), block 256 (8 waves); wave -> 16(M) x 128(N), K=160
__global__ void k2_ygemm(const float* __restrict__ a2, const float* __restrict__ hn,
                         const float* __restrict__ sstart, const float* __restrict__ alpha_logit,
                         float* __restrict__ mix) {
    const int nslice = blockIdx.x, blk = blockIdx.y, b = blockIdx.z;
    const int t = threadIdx.x, wave = t >> 5, lane = t & 31;
    const int half = lane >> 4, lid = lane & 15;
    const float alpha = sigmoidf_(alpha_logit[0]);
    const float* A  = a2 + ((size_t)(b * NBLK + blk)) * 128 * 160;
    const float* hb = hn + ((size_t)b * Ww + blk * BLK) * Dd;
    const float* sb = sstart + ((size_t)(b * NBLK + blk)) * Rr * Dd;
    const int n_base = nslice * 128;
    const int arow = wave * 16 + lid;
    v8f acc[8];
#pragma unroll
    for (int i = 0; i < 8; ++i) acc[i] = (v8f){0,0,0,0,0,0,0,0};
    for (int k = 0; k < 160; k += 4) {
        const int ka = k + 2 * half;
        v2f a;
        a.x = A[(size_t)arow * 160 + ka];
        a.y = A[(size_t)arow * 160 + ka + 1];
        const float* br0;
        const float* br1;
        if (k < 128) { br0 = hb + (size_t)ka * Dd;         br1 = hb + (size_t)(ka + 1) * Dd; }
        else         { br0 = sb + (size_t)(ka - 128) * Dd; br1 = sb + (size_t)(ka - 127) * Dd; }
#pragma unroll
        for (int nt = 0; nt < 8; ++nt) {
            const int n = n_base + nt * 16 + lid;
            v2f bb; bb.x = br0[n]; bb.y = br1[n];
            acc[nt] = WMMA4(a, bb, acc[nt]);
        }
    }
#pragma unroll
    for (int nt = 0; nt < 8; ++nt) {
#pragma unroll
        for (int i = 0; i < 8; ++i) {
            const int m = blk * BLK + wave * 16 + i + 8 * half;
            const int n = n_base + nt * 16 + lid;
            const size_t idx = ((size_t)b * Ww + m) * Dd + n;
            mix[idx] += alpha * acc[nt][i];
        }
    }
}

// ---------------- 7/9/10: generic LDS-staged NT GEMM  C = A @ Bw^T (+bias, +res | gelu) ----
// grid (N/64, M/64), block 256; 8 waves, each 2 C-tiles of 16x16; K chunked by 32.
// float4 global loads + ds_store_b128; row stride 36 keeps b128 alignment + conflict-free reads.
__global__ void k2_nt_gemm(const float* __restrict__ A, const float* __restrict__ Bw,
                           const float* __restrict__ bias, const float* __restrict__ res,
                           float* __restrict__ out, int M, int N, int K, int mode) {
    __shared__ float As[64][36];
    __shared__ float Bs[64][36];
    const int t = threadIdx.x, wave = t >> 5, lane = t & 31;
    const int half = lane >> 4, lid = lane & 15;
    const int m_base = blockIdx.y * 64, n_base = blockIdx.x * 64;
    const int mt = wave & 3, nt0 = wave >> 2;            // tiles (mt,nt0) and (mt,nt0+2)
    v8f acc0 = {0,0,0,0,0,0,0,0}, acc1 = acc0;
    for (int k0 = 0; k0 < K; k0 += 32) {
        __syncthreads();
        for (int i = t; i < 64 * 8; i += 256) {          // 512 float4s per matrix
            const int r = i >> 3, c4 = (i & 7) * 4;
            const float4 av = *(const float4*)(A  + (size_t)(m_base + r) * K + k0 + c4);
            const float4 bv = *(const float4*)(Bw + (size_t)(n_base + r) * K + k0 + c4);
            *(float4*)&As[r][c4] = av;
            *(float4*)&Bs[r][c4] = bv;
        }
        __syncthreads();
#pragma unroll
        for (int k = 0; k < 32; k += 4) {
            const int ka = k + 2 * half;
            v2f a, b0, b1;
            a.x  = As[mt * 16 + lid][ka];        a.y  = As[mt * 16 + lid][ka + 1];
            b0.x = Bs[nt0 * 16 + lid][ka];       b0.y = Bs[nt0 * 16 + lid][ka + 1];
            b1.x = Bs[(nt0 + 2) * 16 + lid][ka]; b1.y = Bs[(nt0 + 2) * 16 + lid][ka + 1];
            acc0 = WMMA4(a, b0, acc0);
            acc1 = WMMA4(a, b1, acc1);
        }
    }
#pragma unroll
    for (int i = 0; i < 8; ++i) {
        const int m = m_base + mt * 16 + i + 8 * half;
        const int n0 = n_base + nt0 * 16 + lid;
        const int n1 = n_base + (nt0 + 2) * 16 + lid;
        float v0 = acc0[i] + bias[n0];
        float v1 = acc1[i] + bias[n1];
        if (mode == 0) {
            v0 += res[(size_t)m * N + n0];
            v1 += res[(size_t)m * N + n1];
        } else {
            v0 = geluf_(v0);
            v1 = geluf_(v1);
        }
        out[(size_t)m * N + n0] = v0;
        out[(size_t)m * N + n1] = v1;
    }
}

// ---------------- launcher ----------------
extern "C" void kernel_launch(void* const* d_in, const int* in_sizes, int n_in,
                              void* d_out, int out_size, void* d_ws, size_t ws_size,
                              hipStream_t stream) {
    const float* h           = (const float*)d_in[0];
    const float* k_base      = (const float*)d_in[1];
    const float* decay_logit = (const float*)d_in[2];
    const float* gate_logit  = (const float*)d_in[3];
    const float* u           = (const float*)d_in[4];
    const float* v           = (const float*)d_in[5];
    const float* alpha_logit = (const float*)d_in[6];
    const float* proj_w      = (const float*)d_in[7];
    const float* proj_b      = (const float*)d_in[8];
    const float* norm1_scale = (const float*)d_in[9];
    const float* norm2_scale = (const float*)d_in[10];
    const float* up_w        = (const float*)d_in[11];
    const float* up_b        = (const float*)d_in[12];
    const float* down_w      = (const float*)d_in[13];
    const float* down_b      = (const float*)d_in[14];
    float* out = (float*)d_out;

    float* ws = (float*)d_ws;
    const size_t SZ_ROWD = (size_t)ROWS * Dd;              // 4M floats
    float* hn     = ws;                                    // also reused as mnorm later
    float* mix    = hn + SZ_ROWD;
    float* h2     = mix + SZ_ROWD;
    float* up_out = h2 + SZ_ROWD;                          // 8M floats
    float* qbuf   = up_out + 2 * SZ_ROWD;                  // ROWS*R
    float* kbuf   = qbuf + (size_t)ROWS * Rr;
    float* cblk   = kbuf + (size_t)ROWS * Rr;              // B*NBLK*R*D = 1M
    float* sstart = cblk + (size_t)Bb * NBLK * Rr * Dd;    // 1M
    float* a2     = sstart + (size_t)Bb * NBLK * Rr * Dd;  // B*NBLK*128*160

    // 1. h_norm = rmsnorm(h, norm1_scale)
    k2_rmsnorm<<<ROWS, 256, 0, stream>>>(h, norm1_scale, hn);
    // 2. q, k (l2-normalized low-rank projections)
    k2_qk<<<ROWS, 128, 0, stream>>>(hn, u, v, qbuf, kbuf);
    // 3. base causal mix -> mix
    k2_base_gemm<<<dim3(2, Ww / 16, Bb), 256, 0, stream>>>(k_base, hn, gate_logit, mix);
    // 4. per-block decayed contributions
    k2_cblk<<<dim3(Rr / 2, NBLK, Bb), 256, 0, stream>>>(hn, kbuf, decay_logit, cblk);
    // 5. sequential block-state scan
    k2_scan<<<(Bb * Rr * Dd) / 256, 256, 0, stream>>>(cblk, decay_logit, sstart);
    // 6a. decayed causal scores + q' -> packed [128 x 160] operand
    k2_scores<<<dim3(NBLK, Bb), 256, 0, stream>>>(qbuf, kbuf, decay_logit, a2);
    // 6b. y GEMM, mix += alpha * y
    k2_ygemm<<<dim3(Dd / 128, NBLK, Bb), 256, 0, stream>>>(a2, hn, sstart, alpha_logit, mix);
    // 7. h2 = h + mix @ proj_w^T + proj_b
    k2_nt_gemm<<<dim3(Dd / 64, ROWS / 64), 256, 0, stream>>>(mix, proj_w, proj_b, h, h2,
                                                             ROWS, Dd, Dd, 0);
    // 8. mnorm = rmsnorm(h2, norm2_scale)   (reuse hn buffer)
    k2_rmsnorm<<<ROWS, 256, 0, stream>>>(h2, norm2_scale, hn);
    // 9. up_out = gelu(mnorm @ up_w^T + up_b)
    k2_nt_gemm<<<dim3((2 * Dd) / 64, ROWS / 64), 256, 0, stream>>>(hn, up_w, up_b, nullptr,
                                                                   up_out, ROWS, 2 * Dd, Dd, 1);
    // 10. out = h2 + up_out @ down_w^T + down_b
    k2_nt_gemm<<<dim3(Dd / 64, ROWS / 64), 256, 0, stream>>>(up_out, down_w, down_b, h2, out,
                                                             ROWS, Dd, 2 * Dd, 0);
    (void)in_sizes; (void)n_in; (void)out_size; (void)ws_size;
}